// StaticHyperGRU_35751307772220
// MI455X (gfx1250) — compile-verified
//
#include <hip/hip_runtime.h>
#include <hip/hip_bf16.h>

// ---------------------------------------------------------------------------
// StaticHyperGRU for MI455X (gfx1250, wave32)
//   Phase 0: cond MLP (tiny, scalar)                        -> ws_w [64x8]
//   Phase 1: hyper projections as [64x8]@[8x912] GEMM via
//            V_WMMA_F32_16X16X4_F32 (4 x 57 tiles, K=8 -> 2 chained WMMAs)
//   Phase 2: latency-bound GRU scan, one wave32 per sample:
//            - h state in SGPRs (v_readlane broadcast)
//            - split-K FMA trees (8-deep, 4 independent chains)
//            - v_permlanex16 half-swap for inputgate (VALU, not LDS)
//            - double-buffered x chunk loads
//            - y projection via ds_swizzle xor tree (off critical path)
// ---------------------------------------------------------------------------

#define B_   64
#define T_   32768
#define R_   16
#define G3R  48      // G * R
#define MLP_ 8

typedef float v2f __attribute__((ext_vector_type(2)));
typedef float v8f __attribute__((ext_vector_type(8)));

// ds_swizzle wrappers (offset must be an immediate)
#define SWZ(v, imm) __uint_as_float((unsigned)__builtin_amdgcn_ds_swizzle((int)__float_as_uint(v), (imm)))
#define SWZ_X1  0x041F
#define SWZ_X2  0x081F
#define SWZ_X4  0x101F
#define SWZ_X8  0x201F
#define SWZ_X16 0x401F   // swap 16-lane halves

// VALU half-swap: lane j<16 <- lane 16+j, lane 16+j <- lane j (identity selects)
__device__ __forceinline__ float halfswap(float v) {
  int i = (int)__float_as_uint(v);
  int r = __builtin_amdgcn_permlanex16(i, i, 0x76543210, 0xFEDCBA98, false, false);
  return __uint_as_float((unsigned)r);
}

__device__ __forceinline__ float rdlane(float v, int l) {
  return __uint_as_float((unsigned)__builtin_amdgcn_readlane((int)__float_as_uint(v), l));
}

__device__ __forceinline__ float fsig(float v) {
  // 1 / (1 + exp(-v)) using v_exp_f32 (exp2) + v_rcp_f32
  float e = __builtin_amdgcn_exp2f(v * -1.44269504088896340736f);
  return __builtin_amdgcn_rcpf(1.0f + e);
}

__device__ __forceinline__ float ftanh(float v) {
  // tanh(v) = 1 - 2/(exp(2v)+1)
  float e = __builtin_amdgcn_exp2f(v * 2.88539008177792681472f);
  return 1.0f - 2.0f * __builtin_amdgcn_rcpf(e + 1.0f);
}

__device__ __forceinline__ float leaky(float v) { return v >= 0.0f ? v : 0.1f * v; }

// ---------------------------------------------------------------------------
// Phase 0: w = leaky(leaky(c @ W1^T + b1) @ W2^T + b2)   [64 x 8]
// ---------------------------------------------------------------------------
__global__ __launch_bounds__(64) void hyper_mlp_kernel(
    const float* __restrict__ c, const float* __restrict__ w1, const float* __restrict__ b1,
    const float* __restrict__ w2, const float* __restrict__ b2, float* __restrict__ wout) {
  int b = threadIdx.x;
  if (b >= B_) return;
  float h1[MLP_];
#pragma unroll
  for (int m = 0; m < MLP_; ++m) {
    float v = b1[m];
#pragma unroll
    for (int k = 0; k < MLP_; ++k) v = fmaf(c[b * MLP_ + k], w1[m * MLP_ + k], v);
    h1[m] = leaky(v);
  }
#pragma unroll
  for (int m = 0; m < MLP_; ++m) {
    float v = b2[m];
#pragma unroll
    for (int k = 0; k < MLP_; ++k) v = fmaf(h1[k], w2[m * MLP_ + k], v);
    wout[b * MLP_ + m] = leaky(v);
  }
}

// ---------------------------------------------------------------------------
// Phase 1: [64x8] @ [8x912] + bias, tiled 16x16, via v_wmma_f32_16x16x4_f32.
//   grid = (4, 57), block = 32 (one wave per tile).
//   Column tiles: [0,48) -> w_hh(768), [48,51) -> w_ih(48),
//                 [51,54) -> b_ih(48), [54,57) -> b_hh(48)
// ---------------------------------------------------------------------------
__global__ __launch_bounds__(32) void hyper_proj_kernel(
    const float* __restrict__ wbuf,
    const float* __restrict__ phh_w, const float* __restrict__ phh_b,
    const float* __restrict__ pih_w, const float* __restrict__ pih_b,
    const float* __restrict__ pbih_w, const float* __restrict__ pbih_b,
    const float* __restrict__ pbhh_w, const float* __restrict__ pbhh_b,
    float* __restrict__ whh, float* __restrict__ wih,
    float* __restrict__ bih, float* __restrict__ bhh) {
  const int lane = threadIdx.x;
  const int tm = blockIdx.x;   // batch tile: 0..3
  const int tn = blockIdx.y;   // feature tile: 0..56

  const float* pw;
  const float* pb;
  float* dst;
  int dstride, rowbase;
  if (tn < 48)      { pw = phh_w;  pb = phh_b;  rowbase = tn * 16;        dst = whh; dstride = 768; }
  else if (tn < 51) { pw = pih_w;  pb = pih_b;  rowbase = (tn - 48) * 16; dst = wih; dstride = G3R; }
  else if (tn < 54) { pw = pbih_w; pb = pbih_b; rowbase = (tn - 51) * 16; dst = bih; dstride = G3R; }
  else              { pw = pbhh_w; pb = pbhh_b; rowbase = (tn - 54) * 16; dst = bhh; dstride = G3R; }

  const int half = lane >> 4;      // 0 | 1
  const int idx  = lane & 15;
  const int fr   = rowbase + idx;  // global feature row (N index)
  const int brow = tm * 16 + idx;  // batch row (M index)
  const int kb   = half * 2;       // K base for this half-wave

  v2f a0, a1, b0, b1;
  a0.x = wbuf[brow * MLP_ + kb];     a0.y = wbuf[brow * MLP_ + kb + 1];
  a1.x = wbuf[brow * MLP_ + kb + 4]; a1.y = wbuf[brow * MLP_ + kb + 5];
  b0.x = pw[fr * MLP_ + kb];         b0.y = pw[fr * MLP_ + kb + 1];
  b1.x = pw[fr * MLP_ + kb + 4];     b1.y = pw[fr * MLP_ + kb + 5];

  v8f acc = {0.f, 0.f, 0.f, 0.f, 0.f, 0.f, 0.f, 0.f};
  acc = __builtin_amdgcn_wmma_f32_16x16x4_f32(false, a0, false, b0, (short)0, acc, false, false);
  acc = __builtin_amdgcn_wmma_f32_16x16x4_f32(false, a1, false, b1, (short)0, acc, false, false);

  const float bias = pb[fr];
#pragma unroll
  for (int v = 0; v < 8; ++v) {
    int row = tm * 16 + v + half * 8;
    dst[row * dstride + fr] = acc[v] + bias;
  }
}

// ---------------------------------------------------------------------------
// Phase 2: GRU scan. One wave32 per batch sample; 32768 sequential steps.
//   lanes 0..15 : reset-gate col j (colA=j)   and new-gate col j (colB=32+j)
//   lanes 16..31: input-gate col j (colA=16+j)
// ---------------------------------------------------------------------------
__global__ __launch_bounds__(32) void gru_scan_kernel(
    const float* __restrict__ x, const float* __restrict__ h0,
    const float* __restrict__ whh, const float* __restrict__ wih,
    const float* __restrict__ bih, const float* __restrict__ bhh,
    const float* __restrict__ out_w, const float* __restrict__ out_b,
    float* __restrict__ y, float* __restrict__ hout) {
  const int lane = threadIdx.x;
  const int b = blockIdx.x;
  const bool lo = lane < 16;
  const int j = lane & 15;
  const int colA = lane;                 // r-gate (lo) / i-gate (hi)
  const int colB = lo ? (32 + j) : lane; // n-gate (lo) / duplicate (hi, unused)

  const float* W = whh + b * 768;
  float WA[R_], WB[R_];
#pragma unroll
  for (int r = 0; r < R_; ++r) {
    WA[r] = W[r * G3R + colA];
    WB[r] = W[r * G3R + colB];
  }
  const float wihA = wih[b * G3R + colA], wihB = wih[b * G3R + colB];
  const float bihA = bih[b * G3R + colA], bihB = bih[b * G3R + colB];
  const float bhhA = bhh[b * G3R + colA], bhhB = bhh[b * G3R + colB];
  const float owl = lo ? out_w[j] : 0.0f;
  const float ob = out_b[0];

  float hlane = lo ? h0[b * R_ + j] : 0.0f;
  float hsc[R_];   // uniform (SGPR) copies of h
#pragma unroll
  for (int r = 0; r < R_; ++r) hsc[r] = rdlane(hlane, r);

  const float* xb = x + (size_t)b * T_;
  float* yb = y + (size_t)b * T_;

  float xcur = xb[lane];   // first chunk
  for (int t0 = 0; t0 < T_; t0 += 32) {
    // prefetch next chunk; consumed next iteration (hides HBM/L2 latency)
    const float xnext = (t0 + 32 < T_) ? xb[t0 + 32 + lane] : 0.0f;
    float ybuf = 0.0f;
#pragma unroll
    for (int i = 0; i < 32; ++i) {
      const float xt = rdlane(xcur, i);                // uniform x[t]
      const float ihA = fmaf(xt, wihA, bihA);
      const float ihB = fmaf(xt, wihB, bihB);
      // split-K dot products: 4 independent 8-deep FMA chains
      float hhA0 = bhhA, hhA1 = 0.0f, hhB0 = bhhB, hhB1 = 0.0f;
#pragma unroll
      for (int r = 0; r < 8; ++r) {
        hhA0 = fmaf(hsc[r],     WA[r],     hhA0);
        hhA1 = fmaf(hsc[r + 8], WA[r + 8], hhA1);
        hhB0 = fmaf(hsc[r],     WB[r],     hhB0);
        hhB1 = fmaf(hsc[r + 8], WB[r + 8], hhB1);
      }
      const float hhA = hhA0 + hhA1;
      const float hhB = hhB0 + hhB1;
      const float gA = fsig(ihA + hhA);                // reset (lo) / input (hi)
      const float ng = ftanh(fmaf(gA, hhB, ihB));      // newgate (lo lanes)
      const float ig = halfswap(gA);                   // VALU permlanex16 swap
      const float hy = fmaf(ig, hlane - ng, ng);       // GRU blend
      hlane = hy;                                      // hi-lane value is dead
#pragma unroll
      for (int r = 0; r < R_; ++r) hsc[r] = rdlane(hy, r);
      // output projection: xor-swizzle tree (off the h critical path)
      float p = hy * owl;                              // hi lanes contribute 0
      p += SWZ(p, SWZ_X1);
      p += SWZ(p, SWZ_X2);
      p += SWZ(p, SWZ_X4);
      p += SWZ(p, SWZ_X8);
      p += SWZ(p, SWZ_X16);
      const float yt = p + ob;
      if (lane == i) ybuf = yt;                        // v_cndmask select
    }
    yb[t0 + lane] = ybuf;                              // coalesced 32-wide store
    xcur = xnext;
  }
  if (lo) hout[b * R_ + j] = hlane;
}

// ---------------------------------------------------------------------------
// Host launcher
// ---------------------------------------------------------------------------
extern "C" void kernel_launch(void* const* d_in, const int* in_sizes, int n_in,
                              void* d_out, int out_size, void* d_ws, size_t ws_size,
                              hipStream_t stream) {
  const float* x       = (const float*)d_in[0];
  const float* c       = (const float*)d_in[1];
  const float* h0      = (const float*)d_in[2];
  const float* mlp_w1  = (const float*)d_in[3];
  const float* mlp_b1  = (const float*)d_in[4];
  const float* mlp_w2  = (const float*)d_in[5];
  const float* mlp_b2  = (const float*)d_in[6];
  const float* pih_w   = (const float*)d_in[7];
  const float* pih_b   = (const float*)d_in[8];
  const float* phh_w   = (const float*)d_in[9];
  const float* phh_b   = (const float*)d_in[10];
  const float* pbih_w  = (const float*)d_in[11];
  const float* pbih_b  = (const float*)d_in[12];
  const float* pbhh_w  = (const float*)d_in[13];
  const float* pbhh_b  = (const float*)d_in[14];
  const float* out_w   = (const float*)d_in[15];
  const float* out_b   = (const float*)d_in[16];

  float* y    = (float*)d_out;                   // [B, 1, T]
  float* hout = (float*)d_out + (size_t)B_ * T_; // [B, 1, R]

  // workspace layout (floats)
  float* ws      = (float*)d_ws;
  float* ws_w    = ws;                 // 64*8    = 512
  float* ws_whh  = ws_w + 512;         // 64*768  = 49152
  float* ws_wih  = ws_whh + 49152;     // 64*48   = 3072
  float* ws_bih  = ws_wih + 3072;      // 64*48   = 3072
  float* ws_bhh  = ws_bih + 3072;      // 64*48   = 3072

  hyper_mlp_kernel<<<dim3(1), dim3(64), 0, stream>>>(c, mlp_w1, mlp_b1, mlp_w2, mlp_b2, ws_w);

  hyper_proj_kernel<<<dim3(4, 57), dim3(32), 0, stream>>>(
      ws_w, phh_w, phh_b, pih_w, pih_b, pbih_w, pbih_b, pbhh_w, pbhh_b,
      ws_whh, ws_wih, ws_bih, ws_bhh);

  gru_scan_kernel<<<dim3(B_), dim3(32), 0, stream>>>(
      x, h0, ws_whh, ws_wih, ws_bih, ws_bhh, out_w, out_b, y, hout);
}